// RayMarchingExtendedSpaceGenerateInsideLayer_90271622627609
// MI455X (gfx1250) — compile-verified
//
#include <hip/hip_runtime.h>
#include <math.h>

typedef __attribute__((ext_vector_type(2))) float v2f;
typedef __attribute__((ext_vector_type(8))) float v8f;

#define WAVES_PER_BLOCK 8
#define RAYS_PER_WAVE   16
#define HIDDEN          128
#define NTILES          8   // HIDDEN / 16

#define LOG2E 1.4426950408889634f
#define LN2   0.6931471805599453f

// One wave32 processes 16 rays. Lane n and lane n+16 both own ray n (needed to
// build the 16x4 f32 WMMA A-matrix whose VGPR0 holds K=0 in the low half and
// K=2 in the high half). All control flow is uniform so EXEC stays all-ones
// across every v_wmma as the ISA requires.
__global__ __launch_bounds__(256) void raymarch_sdf_kernel(
    const float* __restrict__ r,       // (N, 4): [x0_dir, x, y, z]
    const float* __restrict__ pivot,   // (3,)
    const float* __restrict__ W1,      // (3, 128) row-major
    const float* __restrict__ b1,      // (128,)
    const float* __restrict__ W2,      // (128,)
    const float* __restrict__ b2,      // (1,)
    const int*   __restrict__ n_iter_p,
    float* __restrict__ out,           // (N, 3)
    int nRays)
{
    // Wave-private reduction scratch: [wave][acc-row g][lane]
    __shared__ float red[WAVES_PER_BLOCK][NTILES][32];

    const int tid    = threadIdx.x;
    const int wave   = tid >> 5;
    const int lane   = tid & 31;
    const int ln     = lane & 15;          // ray slot within wave
    const bool hiHalf = (lane >= 16);

    const int rayLogical = (blockIdx.x * WAVES_PER_BLOCK + wave) * RAYS_PER_WAVE + ln;
    const int ray = (rayLogical < nRays) ? rayLogical : (nRays - 1);

    // Load ray direction, normalize over all 4 components (matches reference).
    const float4 rv = ((const float4*)r)[ray];
    const float inv = 1.0f / sqrtf(rv.x*rv.x + rv.y*rv.y + rv.z*rv.z + rv.w*rv.w);
    const float rn0 = rv.x * inv;   // extended coordinate direction
    const float rnx = rv.y * inv;
    const float rny = rv.z * inv;
    const float rnz = rv.w * inv;

    const float px = pivot[0], py = pivot[1], pz = pivot[2];
    const float b2s = b2[0];

    // Persistent B-matrix tiles of W1 (4x16 each): VGPR0 = row K=0 (low half)
    // / row K=2 (high half); VGPR1 = row K=1 / zero pad (K=3).
    float wB0[NTILES], wB1[NTILES], vb1[NTILES], vw2[NTILES];
#pragma unroll
    for (int t = 0; t < NTILES; ++t) {
        const int h = t * 16 + ln;
        wB0[t] = hiHalf ? W1[2 * HIDDEN + h] : W1[0 * HIDDEN + h];
        wB1[t] = hiHalf ? 0.0f               : W1[1 * HIDDEN + h];
        vb1[t] = b1[h];    // hidden index of D element is t*16 + ln for every lane
        vw2[t] = W2[h];
    }

    const int n_iter = *n_iter_p;
    float alpha = 0.0f;

    for (int it = 0; it < n_iter; ++it) {
        // Current point: x = pivot_ext + alpha * r_norm
        const float x0 = alpha * rn0;
        const float xc = px + alpha * rnx;
        const float yc = py + alpha * rny;
        const float zc = pz + alpha * rnz;

        // A-matrix (16 rays x K4): lane n -> (x, y), lane n+16 -> (z, 0)
        v2f amat;
        amat.x = hiHalf ? zc   : xc;
        amat.y = hiHalf ? 0.0f : yc;

        float partial[NTILES];
#pragma unroll
        for (int g = 0; g < NTILES; ++g) partial[g] = 0.0f;

#pragma unroll
        for (int t = 0; t < NTILES; ++t) {
            v2f bmat;
            bmat.x = wB0[t];
            bmat.y = wB1[t];
            v8f c = {0.f, 0.f, 0.f, 0.f, 0.f, 0.f, 0.f, 0.f};
            // D(16 rays x 16 hidden) = A x B ; element (g,lane): ray g+8*hi, hidden t*16+ln
            v8f d = __builtin_amdgcn_wmma_f32_16x16x4_f32(
                false, amat, false, bmat, (short)0, c, false, false);
#pragma unroll
            for (int g = 0; g < 8; ++g) {
                const float pre = d[g] + vb1[t];
                // Numerically stable softplus on raw HW transcendentals:
                //   softplus(x) = max(x,0) + ln2 * log2(1 + 2^(-|x|*log2e))
                // The log2 argument lies in [1,2] -- never subnormal/inf -- so
                // the raw v_exp_f32/v_log_f32 ops are safe (no libm fixups).
                const float e  = __builtin_amdgcn_exp2f(-LOG2E * fabsf(pre));
                const float lg = __builtin_amdgcn_logf(1.0f + e);
                const float sp = fmaf(LN2, lg, fmaxf(pre, 0.0f));
                partial[g] = fmaf(sp, vw2[t], partial[g]);
            }
        }

        // partial[g] @ lane l belongs to ray (g + 8*hi(l)), summed over hidden
        // congruent to ln mod 16. Reduce the 16 lanes of the owning half via a
        // wave-private LDS strip (in-order DS within a wave).
#pragma unroll
        for (int g = 0; g < NTILES; ++g) red[wave][g][lane] = partial[g];
        __builtin_amdgcn_wave_barrier();
        asm volatile("s_wait_dscnt 0" ::: "memory");

        const int gsel = ln & 7;
        const int base = (ln >> 3) * 16;   // which half of the wave holds my ray
        // 64B-aligned: four ds_load_b128
        const float4* rowq = (const float4*)&red[wave][gsel][base];
        const float4 q0 = rowq[0], q1 = rowq[1], q2 = rowq[2], q3 = rowq[3];
        __builtin_amdgcn_wave_barrier();
        float s = b2s
                + ((q0.x + q0.y) + (q0.z + q0.w))
                + ((q1.x + q1.y) + (q1.z + q1.w))
                + ((q2.x + q2.y) + (q2.z + q2.w))
                + ((q3.x + q3.y) + (q3.z + q3.w));

        // ext_sdf and alpha update (branch-free, uniform)
        const float aAbs = fabsf(s);
        float ext = fmaxf(s, x0 - aAbs);
        ext = fmaxf(ext, -aAbs - x0);
        alpha = alpha - ext;
    }

    // Output spatial coordinates; low half writes (high half is a duplicate).
    if (!hiHalf && rayLogical < nRays) {
        out[ray * 3 + 0] = px + alpha * rnx;
        out[ray * 3 + 1] = py + alpha * rny;
        out[ray * 3 + 2] = pz + alpha * rnz;
    }
}

extern "C" void kernel_launch(void* const* d_in, const int* in_sizes, int n_in,
                              void* d_out, int out_size, void* d_ws, size_t ws_size,
                              hipStream_t stream) {
    const float* r     = (const float*)d_in[0];
    const float* pivot = (const float*)d_in[1];
    const float* W1    = (const float*)d_in[2];
    const float* b1    = (const float*)d_in[3];
    const float* W2    = (const float*)d_in[4];
    const float* b2    = (const float*)d_in[5];
    const int*   n_it  = (const int*)d_in[6];
    float* out = (float*)d_out;

    const int nRays = in_sizes[0] / 4;
    const int raysPerBlock = WAVES_PER_BLOCK * RAYS_PER_WAVE;  // 128
    const int blocks = (nRays + raysPerBlock - 1) / raysPerBlock;

    raymarch_sdf_kernel<<<blocks, 256, 0, stream>>>(
        r, pivot, W1, b1, W2, b2, n_it, out, nRays);
}